// DIWDataAug_4939212390573
// MI455X (gfx1250) — compile-verified
//
#include <hip/hip_runtime.h>
#include <hip/hip_bf16.h>

// ---------------------------------------------------------------------------
// Problem constants (from reference): B=64, H=W=448, bg is (H+40)x(W+40),
// OUT_IMG=256, OUT_MSK=64.
// ---------------------------------------------------------------------------
#define H_IN   448
#define W_IN   448
#define BG_W   488
#define OUT_IMG 256
#define OUT_MSK 64

typedef __attribute__((ext_vector_type(2))) float v2f;
typedef __attribute__((ext_vector_type(8))) float v8f;

__device__ __forceinline__ float sat01(float x) {
    return fminf(fmaxf(x, 0.0f), 1.0f);
}

// _src_coords from reference
__device__ __forceinline__ void src_coords(int i, int n_out, int size,
                                           int& t0, int& t1, float& w) {
    float sizef = (float)size;
    float t = ((float)i + 0.5f) * (sizef / (float)n_out) - 0.5f;
    t = fminf(fmaxf(t, 0.0f), sizef - 1.0f);
    float tf = floorf(t);
    t0 = (int)tf;
    t1 = min(t0 + 1, size - 1);
    w  = t - tf;
}

// ---------------------------------------------------------------------------
// Kernel 1: per-sample bounding box of (ms > 0.5). One block per sample.
// ---------------------------------------------------------------------------
__global__ __launch_bounds__(256) void diw_bbox_kernel(
    const float* __restrict__ ms, int* __restrict__ bbox)
{
    const int s   = blockIdx.x;
    const int tid = threadIdx.x;
    const float4* m4 = (const float4*)(ms + (size_t)s * (H_IN * W_IN));
    const int N4 = (H_IN * W_IN) / 4;   // 50176, W divisible by 4

    int lminy = H_IN, lmaxy = -1, lminx = W_IN, lmaxx = -1;

    for (int i = tid; i < N4; i += 256) {
        __builtin_prefetch(&m4[i + 4096], 0, 0);   // global_prefetch_b8
        float4 v = m4[i];
        int base = i * 4;
        int y = base / W_IN;
        int x = base - y * W_IN;
        bool a0 = v.x > 0.5f, a1 = v.y > 0.5f, a2 = v.z > 0.5f, a3 = v.w > 0.5f;
        if (a0 | a1 | a2 | a3) { lminy = min(lminy, y); lmaxy = max(lmaxy, y); }
        if (a0) { lminx = min(lminx, x    ); lmaxx = max(lmaxx, x    ); }
        if (a1) { lminx = min(lminx, x + 1); lmaxx = max(lmaxx, x + 1); }
        if (a2) { lminx = min(lminx, x + 2); lmaxx = max(lmaxx, x + 2); }
        if (a3) { lminx = min(lminx, x + 3); lmaxx = max(lmaxx, x + 3); }
    }

    __shared__ int sb[4];
    if (tid == 0) { sb[0] = H_IN; sb[1] = -1; sb[2] = W_IN; sb[3] = -1; }
    __syncthreads();
    atomicMin(&sb[0], lminy); atomicMax(&sb[1], lmaxy);
    atomicMin(&sb[2], lminx); atomicMax(&sb[3], lmaxx);
    __syncthreads();
    if (tid == 0) {
        int* o = bbox + s * 4;
        o[0] = sb[0]; o[1] = sb[1]; o[2] = sb[2]; o[3] = sb[3];
    }
}

// mask_at() from reference: returns 1.0 if inside crop AND mask set
__device__ __forceinline__ float mask_at(const float* __restrict__ msS,
                                         int miny, int minx, int h, int w,
                                         int top, int left, int P, int Q)
{
    bool in_crop = (P >= top) && (P < top + h) && (Q >= left) && (Q < left + w);
    if (!in_crop) return 0.0f;
    int r = min(max(miny + P - top,  0), H_IN - 1);
    int c = min(max(minx + Q - left, 0), W_IN - 1);
    return (msS[r * W_IN + c] > 0.5f) ? 1.0f : 0.0f;
}

// ---------------------------------------------------------------------------
// Kernel 2: mask bilinear resize to 64x64. grid = (16, B), 256 thr.
// ---------------------------------------------------------------------------
__global__ __launch_bounds__(256) void diw_mask_resize_kernel(
    const float* __restrict__ ms, const int* __restrict__ bbox,
    const int* __restrict__ pads, float* __restrict__ mout)
{
    const int s = blockIdx.y;
    const int p = blockIdx.x * 256 + threadIdx.x;   // 0..4095
    const int oy = p >> 6, ox = p & 63;

    const int miny = bbox[s*4+0], maxy = bbox[s*4+1];
    const int minx = bbox[s*4+2], maxx = bbox[s*4+3];
    const int h = maxy - miny + 1, w = maxx - minx + 1;
    const int left = pads[s*4+0], right = pads[s*4+1];
    const int top  = pads[s*4+2], bot   = pads[s*4+3];
    const int Hp = h + top + bot, Wp = w + left + right;

    int y0, y1, x0, x1; float wy, wx;
    src_coords(oy, OUT_MSK, Hp, y0, y1, wy);
    src_coords(ox, OUT_MSK, Wp, x0, x1, wx);

    const float* msS = ms + (size_t)s * (H_IN * W_IN);
    float m00 = mask_at(msS, miny, minx, h, w, top, left, y0, x0);
    float m01 = mask_at(msS, miny, minx, h, w, top, left, y0, x1);
    float m10 = mask_at(msS, miny, minx, h, w, top, left, y1, x0);
    float m11 = mask_at(msS, miny, minx, h, w, top, left, y1, x1);

    float tr = m00 * (1.0f - wx) + m01 * wx;
    float br = m10 * (1.0f - wx) + m11 * wx;
    mout[(size_t)s * (OUT_MSK * OUT_MSK) + p] = tr * (1.0f - wy) + br * wy;
}

// ---------------------------------------------------------------------------
// Kernel 3: crop/pad/composite + bilinear to 256x256 + brightness clip,
// plus per-block gray partial sums. grid = (256, B), 256 thr (1 px/thr).
// ---------------------------------------------------------------------------
__global__ __launch_bounds__(256) void diw_img_resample_kernel(
    const float* __restrict__ img, const float* __restrict__ ms,
    const float* __restrict__ bg,  const int* __restrict__ bbox,
    const int* __restrict__ pads,  const int* __restrict__ csel,
    const float* __restrict__ color, const float* __restrict__ jitter,
    float* __restrict__ imgOut, float* __restrict__ partials)
{
    const int s = blockIdx.y;
    const int p = blockIdx.x * 256 + threadIdx.x;   // 0..65535
    const int oy = p >> 8, ox = p & 255;

    const int miny = bbox[s*4+0], maxy = bbox[s*4+1];
    const int minx = bbox[s*4+2], maxx = bbox[s*4+3];
    const int h = maxy - miny + 1, w = maxx - minx + 1;
    const int left = pads[s*4+0], right = pads[s*4+1];
    const int top  = pads[s*4+2], bot   = pads[s*4+3];
    const int Hp = h + top + bot, Wp = w + left + right;
    const int sel = csel[s];

    int y0, y1, x0, x1; float wy, wx;
    src_coords(oy, OUT_IMG, Hp, y0, y1, wy);
    src_coords(ox, OUT_IMG, Wp, x0, x1, wx);

    const float* msS  = ms  + (size_t)s * (H_IN * W_IN);
    const float* imgS = img + (size_t)s * 3 * (H_IN * W_IN);
    const float* bgS  = bg  + (size_t)s * 3 * (BG_W * BG_W);
    float col[3] = { color[s*3+0], color[s*3+1], color[s*3+2] };

    const int Ps[4] = { y0, y0, y1, y1 };
    const int Qs[4] = { x0, x1, x0, x1 };
    float v[4][3];
    #pragma unroll
    for (int k = 0; k < 4; ++k) {
        const int P = Ps[k], Q = Qs[k];
        bool in_crop = (P >= top) && (P < top + h) && (Q >= left) && (Q < left + w);
        int r = min(max(miny + P - top,  0), H_IN - 1);
        int c = min(max(minx + Q - left, 0), W_IN - 1);
        bool m = in_crop && (msS[r * W_IN + c] > 0.5f);
        #pragma unroll
        for (int ch = 0; ch < 3; ++ch) {
            float bv = (sel > 2) ? bgS[(size_t)ch * (BG_W * BG_W) + P * BG_W + Q]
                                 : col[ch];
            v[k][ch] = m ? imgS[(size_t)ch * (H_IN * W_IN) + r * W_IN + c] : bv;
        }
    }

    const float bfac = 0.9f + 0.2f * jitter[s*4 + 0];
    float* outS = imgOut + (size_t)s * 3 * (OUT_IMG * OUT_IMG);
    const float lum[3] = { 0.299f, 0.587f, 0.114f };
    float gray = 0.0f;
    #pragma unroll
    for (int ch = 0; ch < 3; ++ch) {
        float tr = v[0][ch] * (1.0f - wx) + v[1][ch] * wx;
        float br = v[2][ch] * (1.0f - wx) + v[3][ch] * wx;
        float o  = tr * (1.0f - wy) + br * wy;
        o = sat01(o * bfac);                 // brightness + clip
        outS[(size_t)ch * (OUT_IMG * OUT_IMG) + p] = o;
        gray += lum[ch] * o;
    }

    // block reduction of gray -> one partial per block (wave32 shuffles)
    float gs = gray;
    #pragma unroll
    for (int off = 16; off > 0; off >>= 1)
        gs += __shfl_down(gs, off, 32);
    __shared__ float wsum[8];
    const int wid = threadIdx.x >> 5, lid = threadIdx.x & 31;
    if (lid == 0) wsum[wid] = gs;
    __syncthreads();
    if (threadIdx.x == 0) {
        float t = 0.0f;
        #pragma unroll
        for (int i = 0; i < 8; ++i) t += wsum[i];
        partials[s * 256 + blockIdx.x] = t;
    }
}

// ---------------------------------------------------------------------------
// Kernel 4: WMMA reduction of 256 partials -> per-sample mean.
// D = A(16x4) x Ones(4x16) + C row-sums A; 4 chained v_wmma_f32_16x16x4_f32
// reduce all 256 values into the accumulator. One wave per sample, EXEC all 1.
// ---------------------------------------------------------------------------
__global__ __launch_bounds__(32) void diw_mean_wmma_kernel(
    const float* __restrict__ partials, float* __restrict__ means)
{
    const int s    = blockIdx.x;
    const int lane = threadIdx.x;              // 0..31, wave32
    const float* pp = partials + s * 256;

    v2f ones; ones[0] = 1.0f; ones[1] = 1.0f;  // B = ones(4x16)
    v8f c = {};
    #pragma unroll
    for (int i = 0; i < 4; ++i) {
        v2f a;                                  // any bijection of 64 vals is fine
        a[0] = pp[i * 64 + lane * 2 + 0];
        a[1] = pp[i * 64 + lane * 2 + 1];
        c = __builtin_amdgcn_wmma_f32_16x16x4_f32(
                /*neg_a=*/false, a, /*neg_b=*/false, ones,
                /*c_mod=*/(short)0, c, /*reuse_a=*/false, /*reuse_b=*/false);
    }
    // every column of D is identical; lane l holds rows of one 8-row half
    float t = c[0]+c[1]+c[2]+c[3]+c[4]+c[5]+c[6]+c[7];
    t += __shfl_xor(t, 16, 32);                // combine the two row-halves
    if (lane == 0) means[s] = t * (1.0f / (float)(OUT_IMG * OUT_IMG));
}

// ---------------------------------------------------------------------------
// Kernel 5: contrast + saturation + HSV hue shift (in place on d_out).
// ---------------------------------------------------------------------------
__global__ __launch_bounds__(256) void diw_jitter_kernel(
    const float* __restrict__ jitter, const float* __restrict__ means,
    float* __restrict__ imgOut)
{
    const int s = blockIdx.y;
    const int p = blockIdx.x * 256 + threadIdx.x;   // 0..65535
    float* base = imgOut + (size_t)s * 3 * (OUT_IMG * OUT_IMG);

    float r = base[p];
    float g = base[OUT_IMG * OUT_IMG + p];
    float b = base[2 * OUT_IMG * OUT_IMG + p];

    const float cfac = 0.9f + 0.2f * jitter[s*4 + 1];
    const float sfac = 0.9f + 0.2f * jitter[s*4 + 2];
    const float hd   = -0.1f + 0.2f * jitter[s*4 + 3];
    const float mean = means[s];

    // contrast
    r = sat01(cfac * r + (1.0f - cfac) * mean);
    g = sat01(cfac * g + (1.0f - cfac) * mean);
    b = sat01(cfac * b + (1.0f - cfac) * mean);
    // saturation
    float gray = 0.299f * r + 0.587f * g + 0.114f * b;
    r = sat01(sfac * r + (1.0f - sfac) * gray);
    g = sat01(sfac * g + (1.0f - sfac) * gray);
    b = sat01(sfac * b + (1.0f - sfac) * gray);

    // rgb -> hsv
    float mx = fmaxf(r, fmaxf(g, b));
    float mn = fminf(r, fminf(g, b));
    float d  = mx - mn;
    float ds = (d > 0.0f) ? d : 1.0f;
    float hch;
    if (mx == r)      { hch = fmodf((g - b) / ds, 6.0f); if (hch < 0.0f) hch += 6.0f; }
    else if (mx == g) { hch = (b - r) / ds + 2.0f; }
    else              { hch = (r - g) / ds + 4.0f; }
    hch *= (1.0f / 6.0f);
    if (!(d > 0.0f)) hch = 0.0f;
    float sv = (mx > 0.0f) ? (d / mx) : 0.0f;
    float vv = mx;

    // hue shift, python-style mod 1
    hch += hd;
    hch -= floorf(hch);

    // hsv -> rgb
    float h6 = hch * 6.0f;
    float fi = floorf(h6);
    float f  = h6 - fi;
    float pq = vv * (1.0f - sv);
    float q  = vv * (1.0f - f * sv);
    float t  = vv * (1.0f - (1.0f - f) * sv);
    int im = (int)fi % 6; if (im < 0) im += 6;
    switch (im) {
        case 0: r = vv; g = t;  b = pq; break;
        case 1: r = q;  g = vv; b = pq; break;
        case 2: r = pq; g = vv; b = t;  break;
        case 3: r = pq; g = q;  b = vv; break;
        case 4: r = t;  g = pq; b = vv; break;
        default: r = vv; g = pq; b = q; break;
    }
    base[p]                         = sat01(r);
    base[OUT_IMG * OUT_IMG + p]     = sat01(g);
    base[2 * OUT_IMG * OUT_IMG + p] = sat01(b);
}

// ---------------------------------------------------------------------------
extern "C" void kernel_launch(void* const* d_in, const int* in_sizes, int n_in,
                              void* d_out, int out_size, void* d_ws, size_t ws_size,
                              hipStream_t stream)
{
    const float* img    = (const float*)d_in[0];
    const float* ms     = (const float*)d_in[1];
    const float* bg     = (const float*)d_in[2];
    const int*   pads   = (const int*)d_in[3];
    const int*   csel   = (const int*)d_in[4];
    const float* color  = (const float*)d_in[5];
    const float* jitter = (const float*)d_in[6];

    const int B = in_sizes[4];   // csel has one entry per sample

    float* out    = (float*)d_out;
    float* imgOut = out;                                       // B*3*256*256
    float* mskOut = out + (size_t)B * 3 * OUT_IMG * OUT_IMG;   // B*64*64

    char*  ws       = (char*)d_ws;
    int*   bbox     = (int*)ws;                                // B*4 ints
    float* partials = (float*)(ws + 1024);                     // B*256 floats
    float* means    = (float*)(ws + 1024 + (size_t)B * 256 * sizeof(float));

    diw_bbox_kernel<<<B, 256, 0, stream>>>(ms, bbox);
    diw_mask_resize_kernel<<<dim3(16, B), 256, 0, stream>>>(ms, bbox, pads, mskOut);
    diw_img_resample_kernel<<<dim3(256, B), 256, 0, stream>>>(
        img, ms, bg, bbox, pads, csel, color, jitter, imgOut, partials);
    diw_mean_wmma_kernel<<<B, 32, 0, stream>>>(partials, means);
    diw_jitter_kernel<<<dim3(256, B), 256, 0, stream>>>(jitter, means, imgOut);
}